// SGP4Layer_79431125172683
// MI455X (gfx1250) — compile-verified
//
#include <hip/hip_runtime.h>
#include <math.h>

#define TWOPI_F   6.28318530717958647692f
#define INV2PI_F  0.15915494309189533577f

// Python-style mod into [0, 2pi): x - floor(x/2pi)*2pi (matches jnp.mod's
// floor-division lowering in f32). FMA gives an accurate residual; clamp
// guards boundary rounding.
__device__ __forceinline__ float pymod2pi(float x) {
    float q = floorf(x * INV2PI_F);
    float r = fmaf(-q, TWOPI_F, x);
    if (r < 0.0f)        r += TWOPI_F;
    else if (r >= TWOPI_F) r -= TWOPI_F;
    return r;
}

// ---------------------------------------------------------------------------
// Init kernel: 1 thread, double precision. Computes all time-independent SGP4
// coefficients from the 7 orbital elements and writes 36 floats to C.
// ---------------------------------------------------------------------------
__global__ void sgp4_init_kernel(const float* __restrict__ p, float* __restrict__ C) {
    if (threadIdx.x != 0 || blockIdx.x != 0) return;

    const double MU = 398600.5;
    const double RE = 6378.137;
    const double XKE = 60.0 / sqrt(RE * RE * RE / MU);
    const double J2 = 0.00108262998905;
    const double J3 = -0.00000253215306;
    const double J4 = -0.00000161098761;
    const double J3OJ2 = J3 / J2;
    const double X2O3 = 2.0 / 3.0;

    double no_kozai = (double)p[0];
    double ecco     = (double)p[1];
    double inclo    = (double)p[2];
    double nodeo    = (double)p[3];
    double argpo    = (double)p[4];
    double mo       = (double)p[5];
    double bstar    = (double)p[6];

    // ---------------- initl ----------------
    double eccsq  = ecco * ecco;
    double omeosq = 1.0 - eccsq;
    double rteosq = sqrt(omeosq);
    double cosio  = cos(inclo);
    double cosio2 = cosio * cosio;
    double ak     = pow(XKE / no_kozai, X2O3);
    double d1     = 0.75 * J2 * (3.0 * cosio2 - 1.0) / (rteosq * omeosq);
    double del_   = d1 / (ak * ak);
    double adel   = ak * (1.0 - del_ * del_ - del_ * (1.0 / 3.0 + 134.0 * del_ * del_ / 81.0));
    del_          = d1 / (adel * adel);
    double no_unkozai = no_kozai / (1.0 + del_);
    double ao     = pow(XKE / no_unkozai, X2O3);
    double sinio  = sin(inclo);
    double po     = ao * omeosq;
    double con42  = 1.0 - 5.0 * cosio2;
    double con41  = -con42 - cosio2 - cosio2;
    double posq   = po * po;
    double rp     = ao * (1.0 - ecco);

    // ---------------- sgp4init (near-earth) ----------------
    double ss     = 78.0 / RE + 1.0;
    double qzms2t = pow((120.0 - 78.0) / RE, 4.0);
    double perige = (rp - 1.0) * RE;
    int    isimp  = (rp < (220.0 / RE + 1.0)) ? 1 : 0;
    double sfour_lo = (perige < 98.0) ? 20.0 : (perige - 78.0);
    double sfour    = (perige < 156.0) ? (sfour_lo / RE + 1.0) : ss;
    double qzms24   = (perige < 156.0) ? pow((120.0 - sfour_lo) / RE, 4.0) : qzms2t;

    double pinvsq = 1.0 / posq;
    double tsi    = 1.0 / (ao - sfour);
    double eta    = ao * ecco * tsi;
    double etasq  = eta * eta;
    double eeta   = ecco * eta;
    double psisq  = fabs(1.0 - etasq);
    double coef   = qzms24 * pow(tsi, 4.0);
    double coef1  = coef / pow(psisq, 3.5);
    double cc2 = coef1 * no_unkozai * (ao * (1.0 + 1.5 * etasq + eeta * (4.0 + etasq))
               + 0.375 * J2 * tsi / psisq * con41 * (8.0 + 3.0 * etasq * (8.0 + etasq)));
    double cc1 = bstar * cc2;
    double ecco_safe = (fabs(ecco) > 1.0e-12) ? ecco : 1.0;
    double cc3 = (ecco > 1.0e-4)
               ? (-2.0 * coef * tsi * J3OJ2 * no_unkozai * sinio / ecco_safe) : 0.0;
    double x1mth2 = 1.0 - cosio2;
    double cc4 = 2.0 * no_unkozai * coef1 * ao * omeosq * (
        eta * (2.0 + 0.5 * etasq) + ecco * (0.5 + 2.0 * etasq)
        - J2 * tsi / (ao * psisq) * (
            -3.0 * con41 * (1.0 - 2.0 * eeta + etasq * (1.5 - 0.5 * eeta))
            + 0.75 * x1mth2 * (2.0 * etasq - eeta * (1.0 + etasq)) * cos(2.0 * argpo)));
    double cc5 = 2.0 * coef1 * ao * omeosq * (1.0 + 2.75 * (etasq + eeta) + eeta * etasq);
    double cosio4 = cosio2 * cosio2;
    double temp1  = 1.5 * J2 * pinvsq * no_unkozai;
    double temp2  = 0.5 * temp1 * J2 * pinvsq;
    double temp3  = -0.46875 * J4 * pinvsq * pinvsq * no_unkozai;
    double mdot = no_unkozai + 0.5 * temp1 * rteosq * con41
                + 0.0625 * temp2 * rteosq * (13.0 - 78.0 * cosio2 + 137.0 * cosio4);
    double argpdot = -0.5 * temp1 * con42
                   + 0.0625 * temp2 * (7.0 - 114.0 * cosio2 + 395.0 * cosio4)
                   + temp3 * (3.0 - 36.0 * cosio2 + 49.0 * cosio4);
    double xhdot1  = -temp1 * cosio;
    double nodedot = xhdot1 + (0.5 * temp2 * (4.0 - 19.0 * cosio2)
                   + 2.0 * temp3 * (3.0 - 7.0 * cosio2)) * cosio;
    double omgcof  = bstar * cc3 * cos(argpo);
    double eeta_safe = (fabs(eeta) > 1.0e-30) ? eeta : 1.0;
    double xmcof = (ecco > 1.0e-4) ? (-X2O3 * coef * bstar / eeta_safe) : 0.0;
    double nodecf = 3.5 * omeosq * xhdot1 * cc1;
    double t2cof  = 1.5 * cc1;
    double denom  = (fabs(cosio + 1.0) > 1.5e-12) ? (1.0 + cosio) : 1.5e-12;
    double xlcof  = -0.25 * J3OJ2 * sinio * (3.0 + 5.0 * cosio) / denom;
    double aycof  = -0.5 * J3OJ2 * sinio;
    double wdel   = 1.0 + eta * cos(mo);
    double delmo  = wdel * wdel * wdel;
    double sinmao = sin(mo);
    double x7thm1 = 7.0 * cosio2 - 1.0;

    double cc1sq  = cc1 * cc1;
    double d2c    = 4.0 * ao * tsi * cc1sq;
    double temp_d = d2c * tsi * cc1 / 3.0;
    double d3c    = (17.0 * ao + sfour) * temp_d;
    double d4c    = 0.5 * temp_d * ao * tsi * (221.0 * ao + 31.0 * sfour) * cc1;
    double t3cof  = d2c + 2.0 * cc1sq;
    double t4cof  = 0.25 * (3.0 * d3c + cc1 * (12.0 * d2c + 10.0 * cc1sq));
    double t5cof  = 0.2 * (3.0 * d4c + 12.0 * cc1 * d3c + 6.0 * d2c * d2c
                  + 15.0 * cc1sq * (2.0 * d2c + cc1sq));

    C[0]  = (float)mo;        C[1]  = (float)mdot;      C[2]  = (float)argpo;
    C[3]  = (float)argpdot;   C[4]  = (float)nodeo;     C[5]  = (float)nodedot;
    C[6]  = (float)nodecf;    C[7]  = (float)cc1;       C[8]  = (float)(bstar * cc4);
    C[9]  = (float)t2cof;     C[10] = (float)omgcof;    C[11] = (float)xmcof;
    C[12] = (float)eta;       C[13] = (float)delmo;     C[14] = (float)sinmao;
    C[15] = (float)d2c;       C[16] = (float)d3c;       C[17] = (float)d4c;
    C[18] = (float)(bstar * cc5);                       C[19] = (float)t3cof;
    C[20] = (float)t4cof;     C[21] = (float)t5cof;     C[22] = isimp ? 1.0f : 0.0f;
    C[23] = (float)no_unkozai;C[24] = (float)ecco;      C[25] = (float)aycof;
    C[26] = (float)xlcof;     C[27] = (float)con41;     C[28] = (float)x1mth2;
    C[29] = (float)x7thm1;    C[30] = (float)sinio;     C[31] = (float)cosio;
    C[32] = (float)inclo;     C[33] = (float)ao;        C[34] = (float)XKE;
    C[35] = (float)(RE * XKE / 60.0);
}

// ---------------------------------------------------------------------------
// Propagation kernel: wave32, one element per thread, single body copy.
// out[0 : 3n)  = positions (km) [n,3];  out[3n : 6n) = velocities (km/s) [n,3]
// ---------------------------------------------------------------------------
__global__ void __launch_bounds__(256) sgp4_prop_kernel(
    const float* __restrict__ t_arr, const float* __restrict__ C,
    float* __restrict__ out, int n)
{
    const float mo    = C[0],  mdot  = C[1],  argpo = C[2],  argpdot = C[3];
    const float nodeo = C[4],  nodedot = C[5], nodecf = C[6], cc1 = C[7];
    const float bcc4  = C[8],  t2cof = C[9],  omgcof = C[10], xmcof = C[11];
    const float eta   = C[12], delmo = C[13], sinmao = C[14];
    const float d2c   = C[15], d3c   = C[16], d4c    = C[17], bcc5 = C[18];
    const float t3cof = C[19], t4cof = C[20], t5cof  = C[21], isimp = C[22];
    const float no_unk= C[23], ecco  = C[24], aycof  = C[25], xlcof = C[26];
    const float con41 = C[27], x1mth2= C[28], x7thm1 = C[29];
    const float sinim = C[30], cosim = C[31], inclo  = C[32], ao = C[33];
    const float xke   = C[34], vkmps = C[35];
    const float inv_xke = 1.0f / xke;
    const float j2f   = 0.00108262998905f;
    const float ref   = 6378.137f;

    long i = (long)blockIdx.x * blockDim.x + threadIdx.x;
    if (i >= n) return;
    float t = t_arr[i];

    // ---- secular gravity + drag ----
    float xmdf   = mo + mdot * t;
    float argpdf = argpo + argpdot * t;
    float nodedf = nodeo + nodedot * t;
    float t2     = t * t;
    float nodem  = nodedf + nodecf * t2;
    float tempa  = 1.0f - cc1 * t;
    float tempe  = bcc4 * t;
    float templ  = t2cof * t2;

    float mm, argpm;
    if (isimp != 0.0f) {
        mm = xmdf; argpm = argpdf;
    } else {
        float delomg = omgcof * t;
        float w      = 1.0f + eta * cosf(xmdf);
        float delm   = xmcof * (w * w * w - delmo);
        float tdel   = delomg + delm;
        mm    = xmdf + tdel;
        argpm = argpdf - tdel;
        float t3 = t2 * t, t4 = t3 * t;
        tempa = tempa - d2c * t2 - d3c * t3 - d4c * t4;
        tempe = tempe + bcc5 * (sinf(mm) - sinmao);
        templ = templ + t3cof * t3 + t4 * (t4cof + t * t5cof);
    }

    float am = ao * tempa * tempa;
    float nm = xke / (am * sqrtf(am));
    float em = fmaxf(ecco - tempe, 1.0e-6f);
    mm = mm + no_unk * templ;
    float xlm = mm + argpm + nodem;
    nodem = pymod2pi(nodem);
    argpm = pymod2pi(argpm);
    xlm   = pymod2pi(xlm);
    mm    = pymod2pi(xlm - argpm - nodem);

    float ep = em, argpp = argpm, nodep = nodem, mp = mm;

    // ---- long-period periodics ----
    float sargp, cargp;
    sincosf(argpp, &sargp, &cargp);
    float axnl = ep * cargp;
    float tlp  = 1.0f / (am * (1.0f - ep * ep));
    float aynl = ep * sargp + tlp * aycof;
    float xl   = mp + argpp + nodep + tlp * xlcof * axnl;

    // ---- Kepler: 10 clipped Newton steps ----
    // Native trig + native divide: the iteration is self-correcting (step -> 0
    // at the fixed point); final sin/cos recomputed accurately below.
    float u = pymod2pi(xl - nodep);
    float eo1 = u;
    #pragma unroll 1
    for (int kk = 0; kk < 10; ++kk) {
        float s = __sinf(eo1), c = __cosf(eo1);
        float tem5 = __fdividef(u - aynl * c + axnl * s - eo1,
                                1.0f - c * axnl - s * aynl);
        tem5 = fminf(fmaxf(tem5, -0.95f), 0.95f);
        eo1 += tem5;
    }
    float sineo1, coseo1;
    sincosf(eo1, &sineo1, &coseo1);

    // ---- short-period periodics ----
    float ecose = axnl * coseo1 + aynl * sineo1;
    float esine = axnl * sineo1 - aynl * coseo1;
    float el2   = axnl * axnl + aynl * aynl;
    float pl    = am * (1.0f - el2);
    float rl    = am * (1.0f - ecose);
    float inv_rl = 1.0f / rl;
    float rdotl = sqrtf(am) * esine * inv_rl;
    float rvdotl= sqrtf(pl) * inv_rl;
    float betal = sqrtf(1.0f - el2);
    float tsp   = esine / (1.0f + betal);
    float a_rl  = am * inv_rl;
    float sinu  = a_rl * (sineo1 - aynl - axnl * tsp);
    float cosu  = a_rl * (coseo1 - axnl + aynl * tsp);
    float su    = atan2f(sinu, cosu);
    float sin2u = 2.0f * cosu * sinu;
    float cos2u = 1.0f - 2.0f * sinu * sinu;
    float tp    = 1.0f / pl;
    float t1p   = 0.5f * j2f * tp;
    float t2p   = t1p * tp;
    float mrt   = rl * (1.0f - 1.5f * t2p * betal * con41)
                + 0.5f * t1p * x1mth2 * cos2u;
    su = su - 0.25f * t2p * x7thm1 * sin2u;
    float xnode = nodep + 1.5f * t2p * cosim * sin2u;
    float xinc  = inclo + 1.5f * t2p * cosim * sinim * cos2u;
    float mvt   = rdotl - nm * t1p * x1mth2 * sin2u * inv_xke;
    float rvdot = rvdotl + nm * t1p * (x1mth2 * cos2u + 1.5f * con41) * inv_xke;

    // ---- orientation vectors -> TEME ----
    float sinsu, cossu; sincosf(su,    &sinsu, &cossu);
    float snod,  cnod;  sincosf(xnode, &snod,  &cnod);
    float sini,  cosi;  sincosf(xinc,  &sini,  &cosi);
    float xmx = -snod * cosi, xmy = cnod * cosi;
    float ux = xmx * sinsu + cnod * cossu;
    float uy = xmy * sinsu + snod * cossu;
    float uz = sini * sinsu;
    float vx = xmx * cossu - cnod * sinsu;
    float vy = xmy * cossu - snod * sinsu;
    float vz = sini * cossu;

    float* po = out + i * 3;
    po[0] = mrt * ux * ref;
    po[1] = mrt * uy * ref;
    po[2] = mrt * uz * ref;
    float* vo = out + (long)3 * n + i * 3;
    vo[0] = (mvt * ux + rvdot * vx) * vkmps;
    vo[1] = (mvt * uy + rvdot * vy) * vkmps;
    vo[2] = (mvt * uz + rvdot * vz) * vkmps;
}

extern "C" void kernel_launch(void* const* d_in, const int* in_sizes, int n_in,
                              void* d_out, int out_size, void* d_ws, size_t ws_size,
                              hipStream_t stream) {
    const float* params = (const float*)d_in[0];   // 7 elements
    const float* t_min  = (const float*)d_in[1];   // N times
    int n = in_sizes[1];
    float* C = (float*)d_ws;                       // 36-float coefficient table

    sgp4_init_kernel<<<1, 1, 0, stream>>>(params, C);

    int threads = 256;                             // 8 wave32 per block
    int blocks = (n + threads - 1) / threads;
    sgp4_prop_kernel<<<blocks, threads, 0, stream>>>(t_min, C, (float*)d_out, n);
}